// NConvUNet_8555574853947
// MI455X (gfx1250) — compile-verified
//
#include <hip/hip_runtime.h>
#include <hip/hip_bf16.h>
#include <math.h>

typedef float v2f __attribute__((ext_vector_type(2)));
typedef float v8f __attribute__((ext_vector_type(8)));

#define EPSN 1e-20f

// ---- workspace meta layout (float offsets) ----
// Padded B-matrices [KP x 16] for each conv layer + softplus'd w7 + kernel sums.
static constexpr long B1o  = 0;      // 52*16  = 832   (w1: Cin=1, K=50 pad 52)
static constexpr long B2o  = 832;    // 100*16 = 1600  (w2)
static constexpr long B3o  = 2432;   // 100*16 = 1600  (w3)
static constexpr long B4o  = 4032;   // 72*16  = 1152  (w4)
static constexpr long B5o  = 5184;   // 72*16  = 1152  (w5)
static constexpr long B6o  = 6336;   // 72*16  = 1152  (w6)
static constexpr long SP7o = 7488;   // 2 (softplus'd w7)
static constexpr long SSo  = 7490;   // sums: s1@+0, s2@+2, s3@+4, s4@+6, s5@+8, s6@+10, s7@+12
static constexpr long BUFo = 8192;   // big buffers start here

__device__ __forceinline__ float sp10(float p) {
  float t = 10.f * p;
  return (t > 20.f) ? p : log1pf(expf(t)) * 0.1f;   // softplus(10p)/10
}

// ---------------- weight prep ----------------
// B matrix columns: n=0,1 -> nomin ch0/ch1 (weights on data*conf planes)
//                   n=2,3 -> denom ch0/ch1 (weights on conf planes); n>=4 -> 0
__device__ void buildBP5(const float* w, int CIN, float* BP, float* s, int t, int nth) {
  const int K  = CIN * 50;
  const int KP = (CIN == 1) ? 52 : 100;
  for (int i = t; i < KP * 16; i += nth) BP[i] = 0.f;
  __syncthreads();
  for (int f = t; f < K; f += nth) {
    int plane = f / (CIN * 25);
    int rem   = f - plane * CIN * 25;
    int ch    = rem / 25;
    int tap   = rem - ch * 25;
    for (int o = 0; o < 2; ++o) {
      float v = sp10(w[(o * CIN + ch) * 25 + tap]);
      BP[f * 16 + (plane ? 2 + o : o)] = v;
    }
  }
  if (t < 2) {
    float acc = 0.f;
    for (int i = 0; i < CIN * 25; ++i) acc += sp10(w[t * CIN * 25 + i]);
    s[t] = acc;
  }
  __syncthreads();
}

__device__ void buildBP3(const float* w, float* BP, float* s, int t, int nth) {
  const int K = 72;  // 2 planes * 4ch * 9 taps
  for (int i = t; i < K * 16; i += nth) BP[i] = 0.f;
  __syncthreads();
  for (int f = t; f < K; f += nth) {
    int plane = f / 36;
    int rem   = f - plane * 36;
    int ch    = rem / 9;
    int tap   = rem - ch * 9;
    for (int o = 0; o < 2; ++o) {
      float v = sp10(w[(o * 4 + ch) * 9 + tap]);
      BP[f * 16 + (plane ? 2 + o : o)] = v;
    }
  }
  if (t < 2) {
    float acc = 0.f;
    for (int i = 0; i < 36; ++i) acc += sp10(w[t * 36 + i]);
    s[t] = acc;
  }
  __syncthreads();
}

__global__ void prep_kernel(const float* w1, const float* w2, const float* w3,
                            const float* w4, const float* w5, const float* w6,
                            const float* w7, float* ws) {
  int t = threadIdx.x, nth = blockDim.x;
  buildBP5(w1, 1, ws + B1o, ws + SSo + 0,  t, nth);
  buildBP5(w2, 2, ws + B2o, ws + SSo + 2,  t, nth);
  buildBP5(w3, 2, ws + B3o, ws + SSo + 4,  t, nth);
  buildBP3(w4,    ws + B4o, ws + SSo + 6,  t, nth);
  buildBP3(w5,    ws + B5o, ws + SSo + 8,  t, nth);
  buildBP3(w6,    ws + B6o, ws + SSo + 10, t, nth);
  if (t == 0) {
    float a = sp10(w7[0]), b = sp10(w7[1]);
    ws[SP7o] = a; ws[SP7o + 1] = b; ws[SSo + 12] = a + b;
  }
}

// feature f -> patch ROW offset (constant-foldable after full unroll):
// 5x5: f = plane*(CIN*25) + ch*25 + tap ; patch rows of 20 cols (16 + 4 halo)
template <int CIN>
__host__ __device__ constexpr int featRow5(int f) {
  int plane = f / (CIN * 25);
  int rem   = f - plane * (CIN * 25);
  int ch    = rem / 25;
  int tap   = rem - ch * 25;
  int dy = tap / 5, dx = tap - dy * 5;
  return ((plane * CIN + ch) * 5 + dy) * 20 + dx;
}

// 3x3, 4ch: f = plane*36 + ch*9 + tap ; patch rows of 18 cols (16 + 2 halo)
__host__ __device__ constexpr int featRow3(int f) {
  int plane = f / 36;
  int rem   = f - plane * 36;
  int ch    = rem / 9;
  int tap   = rem - ch * 9;
  int dy = tap / 3, dx = tap - dy * 3;
  return ((plane * 4 + ch) * 3 + dy) * 18 + dx;
}

// ---------------- 5x5 normalized conv (WMMA f32 16x16x4) ----------------
template <int CIN>
__global__ void __launch_bounds__(128)
nconv5_kernel(const float* __restrict__ xin, const float* __restrict__ cin,
              const float* __restrict__ BP,  const float* __restrict__ bias,
              const float* __restrict__ s,
              float* __restrict__ xout, float* __restrict__ cout,
              int Bn, int H, int W) {
  constexpr int K     = CIN * 50;
  constexpr int KP    = (CIN == 1) ? 52 : 100;
  constexpr int P     = 2 * CIN;       // planes: CIN data*conf + CIN conf
  constexpr int PATCH = P * 100;       // P * 5 rows * 20 cols
  __shared__ float shBP[KP * 16];
  __shared__ float shPatch[4][PATCH];
  __shared__ float shRes[4][256];

  const int t = threadIdx.x, wv = t >> 5, lane = t & 31;
  for (int i = t; i < KP * 16; i += 128) shBP[i] = BP[i];
  __syncthreads();

  const int  nsx     = (W + 15) >> 4;
  const long nstrips = (long)Bn * H * nsx;
  long strip = (long)blockIdx.x * 4 + wv;
  if (strip >= nstrips) return;                 // wave-uniform exit (EXEC stays full)
  int  sx  = (int)(strip % nsx);
  long tmp = strip / nsx;
  int  r   = (int)(tmp % H);
  int  b   = (int)(tmp / H);
  int  col0 = sx * 16; if (col0 > W - 16) col0 = W - 16;  // overlap-clamp (idempotent)

  // stage im2col patch: planes [0,CIN)=x*c, [CIN,2CIN)=c ; zero-padded halo
  float* patch = shPatch[wv];
  for (int i = lane; i < PATCH; i += 32) {
    int p = i / 100; int rem = i - p * 100;
    int dy = rem / 20, dx = rem - dy * 20;
    int gr = r + dy - 2, gc = col0 + dx - 2;
    float v = 0.f;
    if (gr >= 0 && gr < H && gc >= 0 && gc < W) {
      int ch = (p < CIN) ? p : p - CIN;
      long base = (((long)b * CIN + ch) * H + gr) * W + gc;
      float cv = cin[base];
      v = (p < CIN) ? xin[base] * cv : cv;
    }
    patch[i] = v;
  }

  // GEMM: D[16px x 16] += A[16px x 4] * B[4 x 16], K chunks of 4.
  // A layout (16x4 f32): lanes 0-15 carry K=kb,kb+1 ; lanes 16-31 K=kb+2,kb+3.
  // All feature->address math folds to constants; one cndmask per operand.
  v8f acc;
#pragma unroll
  for (int k = 0; k < 8; ++k) acc[k] = 0.f;
  const int  n16 = lane & 15;
  const bool hi  = (lane >= 16);
#pragma unroll
  for (int kb = 0; kb < KP; kb += 4) {
    const int f0 = kb + 0, f1 = kb + 1, f2 = kb + 2, f3 = kb + 3;
    // compile-time row offsets (clamped so padded-K reads stay in range)
    const int r0 = featRow5<CIN>(f0 < K ? f0 : 0);
    const int r1 = featRow5<CIN>(f1 < K ? f1 : 0);
    const int r2 = featRow5<CIN>(f2 < K ? f2 : 0);
    const int r3 = featRow5<CIN>(f3 < K ? f3 : 0);
    v2f a, bm;
    float ax = patch[(hi ? r2 : r0) + n16];
    float ay = patch[(hi ? r3 : r1) + n16];
    a.x = ((hi ? f2 : f0) < K) ? ax : 0.f;   // folds to 'ax' except final padded chunk
    a.y = ((hi ? f3 : f1) < K) ? ay : 0.f;
    bm.x = shBP[(hi ? f2 * 16 : f0 * 16) + n16];
    bm.y = shBP[(hi ? f3 * 16 : f1 * 16) + n16];
    acc = __builtin_amdgcn_wmma_f32_16x16x4_f32(false, a, false, bm, (short)0, acc,
                                                false, false);
  }

  // C/D layout: vgpr r holds (M = r + 8*(lane/16), N = lane%16) -> transpose via LDS
  float* res = shRes[wv];
  const int mof = hi ? 8 : 0;
#pragma unroll
  for (int ri = 0; ri < 8; ++ri) res[(ri + mof) * 16 + n16] = acc[ri];

  const int o = hi ? 1 : 0, m = n16;    // lanes 0-15: ch0, lanes 16-31: ch1
  float nom = res[m * 16 + o];
  float den = res[m * 16 + 2 + o];
  long obase = (((long)b * 2 + o) * H + r) * W + col0 + m;
  xout[obase] = nom / (den + EPSN) + bias[o];
  cout[obase] = den / s[o];
}

// ---------- 3x3 nconv with fused nearest-2x upsample + concat ----------
__global__ void __launch_bounds__(128)
nconv3up_kernel(const float* __restrict__ fx, const float* __restrict__ fc,
                const float* __restrict__ cx, const float* __restrict__ cc,
                const float* __restrict__ BP, const float* __restrict__ bias,
                const float* __restrict__ s,
                float* __restrict__ xout, float* __restrict__ cout,
                int Bn, int H, int W, int coarseFirst) {
  constexpr int K     = 72;
  constexpr int PATCH = 8 * 54;         // 8 planes * 3 rows * 18 cols
  __shared__ float shBP[K * 16];
  __shared__ float shPatch[4][PATCH];
  __shared__ float shRes[4][256];

  const int t = threadIdx.x, wv = t >> 5, lane = t & 31;
  for (int i = t; i < K * 16; i += 128) shBP[i] = BP[i];
  __syncthreads();

  const int  nsx     = (W + 15) >> 4;
  const long nstrips = (long)Bn * H * nsx;
  long strip = (long)blockIdx.x * 4 + wv;
  if (strip >= nstrips) return;
  int  sx  = (int)(strip % nsx);
  long tmp = strip / nsx;
  int  r   = (int)(tmp % H);
  int  b   = (int)(tmp / H);
  int  col0 = sx * 16; if (col0 > W - 16) col0 = W - 16;
  const int Hc = H >> 1, Wc = W >> 1;

  float* patch = shPatch[wv];
  for (int i = lane; i < PATCH; i += 32) {
    int p = i / 54; int rem = i - p * 54;
    int dy = rem / 18, dx = rem - dy * 18;
    int gr = r + dy - 1, gc = col0 + dx - 1;
    float v = 0.f;
    if (gr >= 0 && gr < H && gc >= 0 && gc < W) {
      int plane = p >> 2, lch = p & 3;  // lch = logical concat channel
      bool isFine = coarseFirst ? (lch >= 2) : (lch < 2);
      if (isFine) {
        int fch = coarseFirst ? lch - 2 : lch;
        long base = (((long)b * 2 + fch) * H + gr) * W + gc;
        float cv = fc[base];
        v = plane ? cv : fx[base] * cv;
      } else {                          // nearest-2x upsample fused into gather
        int cch = coarseFirst ? lch : lch - 2;
        long base = (((long)b * 2 + cch) * Hc + (gr >> 1)) * Wc + (gc >> 1);
        float cv = cc[base];
        v = plane ? cv : cx[base] * cv;
      }
    }
    patch[i] = v;
  }

  v8f acc;
#pragma unroll
  for (int k = 0; k < 8; ++k) acc[k] = 0.f;
  const int  n16 = lane & 15;
  const bool hi  = (lane >= 16);
#pragma unroll
  for (int kb = 0; kb < K; kb += 4) {
    const int f0 = kb + 0, f1 = kb + 1, f2 = kb + 2, f3 = kb + 3;
    const int r0 = featRow3(f0), r1 = featRow3(f1);
    const int r2 = featRow3(f2), r3 = featRow3(f3);
    v2f a, bm;
    a.x  = patch[(hi ? r2 : r0) + n16];
    a.y  = patch[(hi ? r3 : r1) + n16];
    bm.x = shBP[(hi ? f2 * 16 : f0 * 16) + n16];
    bm.y = shBP[(hi ? f3 * 16 : f1 * 16) + n16];
    acc = __builtin_amdgcn_wmma_f32_16x16x4_f32(false, a, false, bm, (short)0, acc,
                                                false, false);
  }

  float* res = shRes[wv];
  const int mof = hi ? 8 : 0;
#pragma unroll
  for (int ri = 0; ri < 8; ++ri) res[(ri + mof) * 16 + n16] = acc[ri];

  const int o = hi ? 1 : 0, m = n16;
  float nom = res[m * 16 + o];
  float den = res[m * 16 + 2 + o];
  long obase = (((long)b * 2 + o) * H + r) * W + col0 + m;
  xout[obase] = nom / (den + EPSN) + bias[o];
  cout[obase] = den / s[o];
}

// ---------- 2x2 conf-argmax pool (first-max, row-major window order) ----------
__global__ void pool2_kernel(const float* __restrict__ conf, const float* __restrict__ x,
                             float* __restrict__ cds, float* __restrict__ xds,
                             long n, int H, int W) {
  long i = (long)blockIdx.x * blockDim.x + threadIdx.x;
  if (i >= n) return;
  int Wo = W >> 1, Ho = H >> 1;
  int  wo    = (int)(i % Wo);
  long tt    = i / Wo;
  int  ho    = (int)(tt % Ho);
  long plane = tt / Ho;                 // b*2 + ch
  long base = ((long)plane * H + (ho << 1)) * W + (wo << 1);
  float c00 = conf[base], c01 = conf[base + 1];
  float c10 = conf[base + W], c11 = conf[base + W + 1];
  float cm = c00; int idx = 0;
  if (c01 > cm) { cm = c01; idx = 1; }
  if (c10 > cm) { cm = c10; idx = 2; }
  if (c11 > cm) { cm = c11; idx = 3; }
  cds[i] = cm * 0.25f;
  xds[i] = x[base + (idx >> 1) * W + (idx & 1)];
}

// ---------- final 1x1 nconv (2ch -> 1ch), writes xout then cout into d_out ----------
__global__ void nconv1_kernel(const float* __restrict__ x, const float* __restrict__ c,
                              const float* __restrict__ wsp, const float* __restrict__ b7,
                              const float* __restrict__ s7, float* __restrict__ out,
                              long n, int H, int W) {
  long i = (long)blockIdx.x * blockDim.x + threadIdx.x;
  if (i >= n) return;
  long HW = (long)H * W;
  long b = i / HW, pix = i - b * HW;
  long i0 = (b * 2) * HW + pix, i1 = i0 + HW;
  float w0 = wsp[0], w1 = wsp[1];
  float c0v = c[i0], c1v = c[i1];
  float den = w0 * c0v + w1 * c1v;
  float nom = w0 * x[i0] * c0v + w1 * x[i1] * c1v;
  out[i]     = nom / (den + EPSN) + b7[0];
  out[n + i] = den / s7[0];
}

extern "C" void kernel_launch(void* const* d_in, const int* in_sizes, int n_in,
                              void* d_out, int out_size, void* d_ws, size_t ws_size,
                              hipStream_t stream) {
  const float* x0 = (const float*)d_in[0];
  const float* c0 = (const float*)d_in[1];
  const float* w1 = (const float*)d_in[2];  const float* b1 = (const float*)d_in[3];
  const float* w2 = (const float*)d_in[4];  const float* b2 = (const float*)d_in[5];
  const float* w3 = (const float*)d_in[6];  const float* b3 = (const float*)d_in[7];
  const float* w4 = (const float*)d_in[8];  const float* b4 = (const float*)d_in[9];
  const float* w5 = (const float*)d_in[10]; const float* b5 = (const float*)d_in[11];
  const float* w6 = (const float*)d_in[12]; const float* b6 = (const float*)d_in[13];
  const float* w7 = (const float*)d_in[14]; const float* b7 = (const float*)d_in[15];
  float* ws  = (float*)d_ws;
  float* out = (float*)d_out;

  const int Bn = 16;
  const int H0 = 352, W0 = 1216, H1 = 176, W1 = 608, H2 = 88, W2 = 304, H3 = 44, W3 = 152;
  const long N0 = (long)Bn * 2 * H0 * W0;   // 13,697,024 floats per tensor
  const long N1 = (long)Bn * 2 * H1 * W1;
  const long N2 = (long)Bn * 2 * H2 * W2;
  const long N3 = (long)Bn * 2 * H3 * W3;

  long off = BUFo;
  float *XA = ws + off; off += N0;  float *CA = ws + off; off += N0;
  float *XB = ws + off; off += N0;  float *CB = ws + off; off += N0;
  float *P1X = ws + off; off += N1; float *P1C = ws + off; off += N1;
  float *Q1X = ws + off; off += N1; float *Q1C = ws + off; off += N1;
  float *P2X = ws + off; off += N2; float *P2C = ws + off; off += N2;
  float *Q2X = ws + off; off += N2; float *Q2C = ws + off; off += N2;
  float *P3X = ws + off; off += N3; float *P3C = ws + off; off += N3;
  float *Q3X = ws + off; off += N3; float *Q3C = ws + off; off += N3;

  prep_kernel<<<1, 256, 0, stream>>>(w1, w2, w3, w4, w5, w6, w7, ws);

  auto cblocks = [&](int H, int W) {
    long st = (long)Bn * H * ((W + 15) / 16);
    return (int)((st + 3) / 4);
  };

  // encoder (full res)
  nconv5_kernel<1><<<cblocks(H0, W0), 128, 0, stream>>>(x0, c0, ws + B1o, b1, ws + SSo + 0, XA, CA, Bn, H0, W0);
  nconv5_kernel<2><<<cblocks(H0, W0), 128, 0, stream>>>(XA, CA, ws + B2o, b2, ws + SSo + 2, XB, CB, Bn, H0, W0);
  nconv5_kernel<2><<<cblocks(H0, W0), 128, 0, stream>>>(XB, CB, ws + B3o, b3, ws + SSo + 4, XA, CA, Bn, H0, W0); // x1,c1 skip

  pool2_kernel<<<(int)((N1 + 255) / 256), 256, 0, stream>>>(CA, XA, P1C, P1X, N1, H0, W0);
  nconv5_kernel<2><<<cblocks(H1, W1), 128, 0, stream>>>(P1X, P1C, ws + B2o, b2, ws + SSo + 2, Q1X, Q1C, Bn, H1, W1);
  nconv5_kernel<2><<<cblocks(H1, W1), 128, 0, stream>>>(Q1X, Q1C, ws + B3o, b3, ws + SSo + 4, P1X, P1C, Bn, H1, W1); // x2_ds skip

  pool2_kernel<<<(int)((N2 + 255) / 256), 256, 0, stream>>>(P1C, P1X, P2C, P2X, N2, H1, W1);
  nconv5_kernel<2><<<cblocks(H2, W2), 128, 0, stream>>>(P2X, P2C, ws + B2o, b2, ws + SSo + 2, Q2X, Q2C, Bn, H2, W2); // x3_ds skip

  pool2_kernel<<<(int)((N3 + 255) / 256), 256, 0, stream>>>(Q2C, Q2X, P3C, P3X, N3, H2, W2);
  nconv5_kernel<2><<<cblocks(H3, W3), 128, 0, stream>>>(P3X, P3C, ws + B2o, b2, ws + SSo + 2, Q3X, Q3C, Bn, H3, W3); // x4_ds

  // decoder: fused up2 + concat + 3x3 nconv
  nconv3up_kernel<<<cblocks(H2, W2), 128, 0, stream>>>(Q2X, Q2C, Q3X, Q3C, ws + B4o, b4, ws + SSo + 6,  P2X, P2C, Bn, H2, W2, 0); // concat(x3_ds, up(x4))
  nconv3up_kernel<<<cblocks(H1, W1), 128, 0, stream>>>(P1X, P1C, P2X, P2C, ws + B5o, b5, ws + SSo + 8,  Q1X, Q1C, Bn, H1, W1, 0); // concat(x2_ds, up(x34))
  nconv3up_kernel<<<cblocks(H0, W0), 128, 0, stream>>>(XA,  CA,  Q1X, Q1C, ws + B6o, b6, ws + SSo + 10, XB,  CB,  Bn, H0, W0, 1); // concat(up(x23), x1)

  long nOut = (long)Bn * H0 * W0;
  nconv1_kernel<<<(int)((nOut + 255) / 256), 256, 0, stream>>>(XB, CB, ws + SP7o, b7, ws + SSo + 12, out, nOut, H0, W0);

  (void)in_sizes; (void)n_in; (void)out_size; (void)ws_size;
}